// GatedGraphAttentionConv_36455682409141
// MI455X (gfx1250) — compile-verified
//
#include <hip/hip_runtime.h>
#include <hip/hip_bf16.h>

typedef float v2f __attribute__((ext_vector_type(2)));
typedef float v8f __attribute__((ext_vector_type(8)));

#define WSTRIDE 66  // padded LDS row stride (even -> 8B-aligned float2, avoids 64-way bank conflicts)

__device__ __forceinline__ unsigned f2key(float x) {
  unsigned u = __float_as_uint(x);
  return (u & 0x80000000u) ? ~u : (u | 0x80000000u);
}
__device__ __forceinline__ float key2f(unsigned k) {
  unsigned u = (k & 0x80000000u) ? (k & 0x7fffffffu) : ~k;
  return __uint_as_float(u);
}
__device__ __forceinline__ float sigmoidf(float x) { return 1.f / (1.f + expf(-x)); }

// ---------------------------------------------------------------------------
// Node-side 4x GEMM: out[m] = X @ W[m] + b[m]  for m in {sg, dg, su, du}
// One wave32 per 16-row tile, V_WMMA_F32_16X16X4_F32, W transposed in LDS.
// ---------------------------------------------------------------------------
__global__ __launch_bounds__(256) void k_node_gemm4(
    const float* __restrict__ X,
    const float* __restrict__ W0, const float* __restrict__ B0,
    const float* __restrict__ W1, const float* __restrict__ B1,
    const float* __restrict__ W2, const float* __restrict__ B2,
    const float* __restrict__ W3, const float* __restrict__ B3,
    float* __restrict__ O0, float* __restrict__ O1,
    float* __restrict__ O2, float* __restrict__ O3,
    int nrows) {
  __shared__ float wt[64 * WSTRIDE];
  const int lane = threadIdx.x & 31;
  const int wave = threadIdx.x >> 5;
  const int half = lane >> 4;   // 0: K even pair, 1: K odd pair
  const int mrow = lane & 15;
  const int ntiles = nrows >> 4;
  const int tile = blockIdx.x * 8 + wave;
  const bool valid = tile < ntiles;
  const int base = valid ? (tile << 4) : 0;

  // A fragments: lane holds row M=mrow, float2 at k = 4s + 2*half
  v2f a[16];
  const float* xrow = X + (size_t)(base + mrow) * 64 + 2 * half;
#pragma unroll
  for (int s = 0; s < 16; ++s) a[s] = *(const v2f*)(xrow + 4 * s);

  const float* Ws[4] = {W0, W1, W2, W3};
  const float* Bs[4] = {B0, B1, B2, B3};
  float* Os[4] = {O0, O1, O2, O3};

  for (int mat = 0; mat < 4; ++mat) {
    __syncthreads();
    // transpose-preload: wt[n*WSTRIDE + k] = W[k][n]
    for (int idx = threadIdx.x; idx < 4096; idx += 256) {
      int k = idx >> 6, n = idx & 63;
      wt[n * WSTRIDE + k] = Ws[mat][idx];
    }
    __syncthreads();
    const float* bias = Bs[mat];
    float* out = Os[mat];
#pragma unroll
    for (int nc = 0; nc < 4; ++nc) {
      v8f c = {0.f, 0.f, 0.f, 0.f, 0.f, 0.f, 0.f, 0.f};
      const float* wb = &wt[(nc * 16 + mrow) * WSTRIDE + 2 * half];
#pragma unroll
      for (int s = 0; s < 16; ++s) {
        v2f b = *(const v2f*)(wb + 4 * s);
        c = __builtin_amdgcn_wmma_f32_16x16x4_f32(false, a[s], false, b,
                                                  (short)0, c, false, false);
      }
      if (valid) {
        int col = nc * 16 + mrow;
        float bv = bias[col];
#pragma unroll
        for (int j = 0; j < 8; ++j) {
          int row = base + j + 8 * half;
          out[(size_t)row * 64 + col] = c[j] + bv;
        }
      }
    }
  }
}

// ---------------------------------------------------------------------------
// Single GEMM (no bias): feat = H @ W_gat
// ---------------------------------------------------------------------------
__global__ __launch_bounds__(256) void k_gemm1(
    const float* __restrict__ X, const float* __restrict__ W,
    float* __restrict__ O, int nrows) {
  __shared__ float wt[64 * WSTRIDE];
  const int lane = threadIdx.x & 31;
  const int wave = threadIdx.x >> 5;
  const int half = lane >> 4;
  const int mrow = lane & 15;
  const int ntiles = nrows >> 4;
  const int tile = blockIdx.x * 8 + wave;
  const bool valid = tile < ntiles;
  const int base = valid ? (tile << 4) : 0;

  for (int idx = threadIdx.x; idx < 4096; idx += 256) {
    int k = idx >> 6, n = idx & 63;
    wt[n * WSTRIDE + k] = W[idx];
  }
  __syncthreads();

  v2f a[16];
  const float* xrow = X + (size_t)(base + mrow) * 64 + 2 * half;
#pragma unroll
  for (int s = 0; s < 16; ++s) a[s] = *(const v2f*)(xrow + 4 * s);

#pragma unroll
  for (int nc = 0; nc < 4; ++nc) {
    v8f c = {0.f, 0.f, 0.f, 0.f, 0.f, 0.f, 0.f, 0.f};
    const float* wb = &wt[(nc * 16 + mrow) * WSTRIDE + 2 * half];
#pragma unroll
    for (int s = 0; s < 16; ++s) {
      v2f b = *(const v2f*)(wb + 4 * s);
      c = __builtin_amdgcn_wmma_f32_16x16x4_f32(false, a[s], false, b,
                                                (short)0, c, false, false);
    }
    if (valid) {
      int col = nc * 16 + mrow;
#pragma unroll
      for (int j = 0; j < 8; ++j)
        O[(size_t)(base + j + 8 * half) * 64 + col] = c[j];
    }
  }
}

// el/er: per (node, head) dot of feat slice with attn vectors
__global__ void k_elr(const float* __restrict__ feat, const float* __restrict__ al,
                      const float* __restrict__ ar, float* __restrict__ el,
                      float* __restrict__ er, int total) {
  int t = blockIdx.x * 256 + threadIdx.x;
  if (t >= total) return;
  int n = t >> 1, h = t & 1;
  const float* f = feat + (size_t)n * 64 + h * 32;
  const float* a = al + h * 32;
  const float* b = ar + h * 32;
  float s1 = 0.f, s2 = 0.f;
#pragma unroll
  for (int d = 0; d < 32; ++d) {
    float fv = f[d];
    s1 += fv * a[d];
    s2 += fv * b[d];
  }
  el[t] = s1;
  er[t] = s2;
}

// ---------------------------------------------------------------------------
// Edge GEMM + fusion: m = esrc_n[src] + edst_n[dst] + EF @ W_eg + b_eg
// Also: logits = leaky_relu(el[src]+er[dst]); atomic-max lmax keys;
//       per-column BN sums for m.  One wave per 16-edge tile.
// ---------------------------------------------------------------------------
__global__ __launch_bounds__(256) void k_edge_gemm(
    const float* __restrict__ EF, const int* __restrict__ src,
    const int* __restrict__ dst, const float* __restrict__ W,
    const float* __restrict__ Bias, const float* __restrict__ esrc_n,
    const float* __restrict__ edst_n, const float* __restrict__ el,
    const float* __restrict__ er, float* __restrict__ m_out,
    float* __restrict__ logits, unsigned* __restrict__ lmax_keys,
    float* __restrict__ sum_m, float* __restrict__ sum_m2, int nedges) {
  __shared__ float wt[64 * WSTRIDE];
  __shared__ float ls1[64], ls2[64];
  const int lane = threadIdx.x & 31;
  const int wave = threadIdx.x >> 5;
  const int half = lane >> 4;
  const int mrow = lane & 15;
  const int ntiles = nedges >> 4;
  const int tile = blockIdx.x * 8 + wave;
  const bool valid = tile < ntiles;
  const int base = valid ? (tile << 4) : 0;

  for (int idx = threadIdx.x; idx < 4096; idx += 256) {
    int k = idx >> 6, n = idx & 63;
    wt[n * WSTRIDE + k] = W[idx];
  }
  if (threadIdx.x < 64) { ls1[threadIdx.x] = 0.f; ls2[threadIdx.x] = 0.f; }
  __syncthreads();

  v2f a[16];
  const float* xrow = EF + (size_t)(base + mrow) * 64 + 2 * half;
#pragma unroll
  for (int s = 0; s < 16; ++s) a[s] = *(const v2f*)(xrow + 4 * s);
  if (valid && tile + 1 < ntiles)
    __builtin_prefetch(EF + (size_t)(base + 16 + mrow) * 64, 0, 1);

#pragma unroll
  for (int nc = 0; nc < 4; ++nc) {
    v8f c = {0.f, 0.f, 0.f, 0.f, 0.f, 0.f, 0.f, 0.f};
    const float* wb = &wt[(nc * 16 + mrow) * WSTRIDE + 2 * half];
#pragma unroll
    for (int s = 0; s < 16; ++s) {
      v2f b = *(const v2f*)(wb + 4 * s);
      c = __builtin_amdgcn_wmma_f32_16x16x4_f32(false, a[s], false, b,
                                                (short)0, c, false, false);
    }
    if (valid) {
      int col = nc * 16 + mrow;
      float bv = Bias[col];
      float ps = 0.f, ps2 = 0.f;
#pragma unroll
      for (int j = 0; j < 8; ++j) {
        int e = base + j + 8 * half;
        int sid = src[e], did = dst[e];
        float v = c[j] + bv + esrc_n[(size_t)sid * 64 + col]
                            + edst_n[(size_t)did * 64 + col];
        m_out[(size_t)e * 64 + col] = v;
        ps += v;
        ps2 += v * v;
      }
      atomicAdd(&ls1[col], ps);
      atomicAdd(&ls2[col], ps2);
    }
  }

  if (valid) {  // logits: lanes 0-15 head 0, lanes 16-31 head 1
    int e = base + mrow;
    int h = half;
    int sid = src[e], did = dst[e];
    float l = el[sid * 2 + h] + er[did * 2 + h];
    float lr = (l > 0.f) ? l : 0.2f * l;
    logits[e * 2 + h] = lr;
    atomicMax(&lmax_keys[did * 2 + h], f2key(lr));
  }
  __syncthreads();
  if (threadIdx.x < 64) {
    atomicAdd(&sum_m[threadIdx.x], ls1[threadIdx.x]);
    atomicAdd(&sum_m2[threadIdx.x], ls2[threadIdx.x]);
  }
}

// decode ordered uint keys -> lmax (zero-in-degree / non-finite -> 0)
__global__ void k_lmax(const unsigned* __restrict__ keys, float* __restrict__ lmax,
                       int total) {
  int t = blockIdx.x * 256 + threadIdx.x;
  if (t >= total) return;
  unsigned k = keys[t];
  float v = 0.f;
  if (k != 0u) {
    v = key2f(k);
    if (!isfinite(v)) v = 0.f;
  }
  lmax[t] = v;
}

// edge pass 2: sigma sums + exp/denom
__global__ void k_edge_pass2(const float* __restrict__ m, const int* __restrict__ src,
                             const int* __restrict__ dst, const float* __restrict__ Bh,
                             const float* __restrict__ logits,
                             const float* __restrict__ lmax, float* __restrict__ ex,
                             float* __restrict__ denom, float* __restrict__ sum_sigma,
                             float* __restrict__ sum_sigma_h, int total) {
  int t = blockIdx.x * 256 + threadIdx.x;
  if (t >= total) return;
  int e = t >> 6, c = t & 63;
  int sid = src[e], did = dst[e];
  float sg = sigmoidf(m[t]);
  float bh = Bh[(size_t)sid * 64 + c];
  atomicAdd(&sum_sigma[(size_t)did * 64 + c], sg);
  atomicAdd(&sum_sigma_h[(size_t)did * 64 + c], bh * sg);
  if (c < 2) {
    float ev = expf(logits[e * 2 + c] - lmax[did * 2 + c]);
    ex[e * 2 + c] = ev;
    atomicAdd(&denom[did * 2 + c], ev);
  }
}

// edge pass 3: alpha-weighted GAT aggregation
__global__ void k_edge_pass3(const int* __restrict__ src, const int* __restrict__ dst,
                             const float* __restrict__ ex, const float* __restrict__ denom,
                             const float* __restrict__ feat, float* __restrict__ gat,
                             int total) {
  int t = blockIdx.x * 256 + threadIdx.x;
  if (t >= total) return;
  int e = t >> 6, c = t & 63, h = c >> 5;
  int sid = src[e], did = dst[e];
  float alpha = ex[e * 2 + h] / denom[did * 2 + h];
  atomicAdd(&gat[(size_t)did * 64 + c], feat[(size_t)sid * 64 + c] * alpha);
}

// node finalize: x_pre = h_agg + gat_out + b_gat; accumulate BN sums
__global__ __launch_bounds__(256) void k_node_fin(
    const float* __restrict__ ss, const float* __restrict__ ssh,
    const float* __restrict__ gat, const float* __restrict__ bgat,
    float* __restrict__ xout, float* __restrict__ sum_x, float* __restrict__ sum_x2,
    int total) {
  __shared__ float s1[64], s2[64];
  int tid = threadIdx.x;
  if (tid < 64) { s1[tid] = 0.f; s2[tid] = 0.f; }
  __syncthreads();
  int t = blockIdx.x * 256 + tid;
  if (t < total) {
    int c = t & 63;
    float xp = ssh[t] / (ss[t] + 1e-6f) + gat[t] + bgat[c];
    xout[t] = xp;
    atomicAdd(&s1[c], xp);
    atomicAdd(&s2[c], xp * xp);
  }
  __syncthreads();
  if (tid < 64) {
    atomicAdd(&sum_x[tid], s1[tid]);
    atomicAdd(&sum_x2[tid], s2[tid]);
  }
}

// BN affine params from sums (biased variance): params = [scale_n|shift_n|scale_e|shift_e]
__global__ void k_bn(const float* __restrict__ sum_x, const float* __restrict__ sum_x2,
                     const float* __restrict__ sum_m, const float* __restrict__ sum_m2,
                     const float* __restrict__ gn, const float* __restrict__ bn,
                     const float* __restrict__ ge, const float* __restrict__ be,
                     float Nf, float Ef, float* __restrict__ params) {
  int c = threadIdx.x;
  if (c >= 64) return;
  float mn = sum_x[c] / Nf;
  float vr = fmaxf(sum_x2[c] / Nf - mn * mn, 0.f);
  float sc = gn[c] * rsqrtf(vr + 1e-5f);
  params[c] = sc;
  params[64 + c] = bn[c] - mn * sc;
  float me = sum_m[c] / Ef;
  float ve = fmaxf(sum_m2[c] / Ef - me * me, 0.f);
  float se = ge[c] * rsqrtf(ve + 1e-5f);
  params[128 + c] = se;
  params[192 + c] = be[c] - me * se;
}

// in-place: out = residual + silu(out*scale + shift)
__global__ void k_out(float* __restrict__ xy, const float* __restrict__ res,
                      const float* __restrict__ scale, const float* __restrict__ shift,
                      int total) {
  int t = blockIdx.x * 256 + threadIdx.x;
  if (t >= total) return;
  int c = t & 63;
  float v = xy[t] * scale[c] + shift[c];
  xy[t] = res[t] + v * sigmoidf(v);
}

extern "C" void kernel_launch(void* const* d_in, const int* in_sizes, int n_in,
                              void* d_out, int out_size, void* d_ws, size_t ws_size,
                              hipStream_t stream) {
  const float* node_feats = (const float*)d_in[0];
  const float* edge_feats = (const float*)d_in[1];
  const int* src = (const int*)d_in[2];
  const int* dst = (const int*)d_in[3];
  const float* W_sg = (const float*)d_in[4];
  const float* b_sg = (const float*)d_in[5];
  const float* W_dg = (const float*)d_in[6];
  const float* b_dg = (const float*)d_in[7];
  const float* W_eg = (const float*)d_in[8];
  const float* b_eg = (const float*)d_in[9];
  const float* W_su = (const float*)d_in[10];
  const float* b_su = (const float*)d_in[11];
  const float* W_du = (const float*)d_in[12];
  const float* b_du = (const float*)d_in[13];
  const float* W_gat = (const float*)d_in[14];
  const float* b_gat = (const float*)d_in[15];
  const float* attn_l = (const float*)d_in[16];
  const float* attn_r = (const float*)d_in[17];
  const float* bn_n_gamma = (const float*)d_in[18];
  const float* bn_n_beta = (const float*)d_in[19];
  const float* bn_e_gamma = (const float*)d_in[20];
  const float* bn_e_beta = (const float*)d_in[21];

  const int N = in_sizes[0] / 64;
  const int E = in_sizes[2];
  const size_t N64 = (size_t)N * 64;
  const size_t E64 = (size_t)E * 64;

  // outputs: x [N,64] then y [E,64]; y-region doubles as storage for m
  float* x_out = (float*)d_out;
  float* y_out = (float*)d_out + N64;

  // workspace layout
  float* p = (float*)d_ws;
  float* e_src_n = p; p += N64;
  float* e_dst_n = p; p += N64;
  float* h_ws = p;    p += N64;
  float* Bh = p;      p += N64;
  float* feat = p;    p += N64;
  float* el = p;      p += 2 * (size_t)N;
  float* er = p;      p += 2 * (size_t)N;
  float* logits = p;  p += 2 * (size_t)E;
  float* ex = p;      p += 2 * (size_t)E;
  float* lmax = p;    p += 2 * (size_t)N;
  float* zstart = p;  // ---- zero-initialized region ----
  unsigned* lmax_keys = (unsigned*)p; p += 2 * (size_t)N;
  float* denom = p;        p += 2 * (size_t)N;
  float* sum_sigma = p;    p += N64;
  float* sum_sigma_h = p;  p += N64;
  float* gat = p;          p += N64;
  float* sum_m = p;        p += 64;
  float* sum_m2 = p;       p += 64;
  float* sum_x = p;        p += 64;
  float* sum_x2 = p;       p += 64;
  float* params = p;       p += 256;
  size_t zbytes = (size_t)((char*)p - (char*)zstart);
  hipMemsetAsync(zstart, 0, zbytes, stream);

  const int node_tiles = N / 16;                    // 3125
  const int node_blocks = (node_tiles + 7) / 8;     // 8 waves/block
  const int edge_tiles = E / 16;                    // 50000
  const int edge_blocks = (edge_tiles + 7) / 8;

  // 1) node GEMMs (WMMA f32)
  k_node_gemm4<<<node_blocks, 256, 0, stream>>>(
      node_feats, W_sg, b_sg, W_dg, b_dg, W_su, b_su, W_du, b_du,
      e_src_n, e_dst_n, h_ws, Bh, N);
  // 2) feat = h @ W_gat (WMMA f32)
  k_gemm1<<<node_blocks, 256, 0, stream>>>(h_ws, W_gat, feat, N);
  // 3) el/er
  k_elr<<<(2 * N + 255) / 256, 256, 0, stream>>>(feat, attn_l, attn_r, el, er, 2 * N);
  // 4) edge GEMM + gathers + logits + lmax + BN(m) sums (WMMA f32)
  k_edge_gemm<<<edge_blocks, 256, 0, stream>>>(
      edge_feats, src, dst, W_eg, b_eg, e_src_n, e_dst_n, el, er,
      y_out /* m */, logits, lmax_keys, sum_m, sum_m2, E);
  // 5) decode lmax
  k_lmax<<<(2 * N + 255) / 256, 256, 0, stream>>>(lmax_keys, lmax, 2 * N);
  // 6) sigma sums + softmax numerator/denominator
  k_edge_pass2<<<(int)((E64 + 255) / 256), 256, 0, stream>>>(
      y_out, src, dst, Bh, logits, lmax, ex, denom, sum_sigma, sum_sigma_h, (int)E64);
  // 7) GAT aggregation
  k_edge_pass3<<<(int)((E64 + 255) / 256), 256, 0, stream>>>(
      src, dst, ex, denom, feat, gat, (int)E64);
  // 8) node finalize + BN(x) sums (writes x_pre into d_out x-region)
  k_node_fin<<<(int)((N64 + 255) / 256), 256, 0, stream>>>(
      sum_sigma, sum_sigma_h, gat, b_gat, x_out, sum_x, sum_x2, (int)N64);
  // 9) BN affine params
  k_bn<<<1, 64, 0, stream>>>(sum_x, sum_x2, sum_m, sum_m2, bn_n_gamma, bn_n_beta,
                             bn_e_gamma, bn_e_beta, (float)N, (float)E, params);
  // 10) outputs: residual + silu(batchnorm(.)) , in place
  k_out<<<(int)((N64 + 255) / 256), 256, 0, stream>>>(x_out, node_feats, params,
                                                      params + 64, (int)N64);
  k_out<<<(int)((E64 + 255) / 256), 256, 0, stream>>>(y_out, edge_feats, params + 128,
                                                      params + 192, (int)E64);
}